// LinearAutoencoder_48610439856506
// MI455X (gfx1250) — compile-verified
//
#include <hip/hip_runtime.h>

typedef __attribute__((ext_vector_type(2))) float v2f;
typedef __attribute__((ext_vector_type(8))) float v8f;

namespace {
constexpr int T_SEQ  = 2048;
constexpr int H_DIM  = 512;
constexpr int MTILE  = 16;   // batch rows per workgroup (= WMMA M)
constexpr int STRIDE = 532;  // LDS row stride in floats: 4*532 % 256 -> full 64-bank coverage for b64 loads
constexpr int YCOL   = 4;    // column where y (= out[3:]) starts inside an LDS row (keeps b64 loads 8B-aligned)
}

// One workgroup = one 16-row batch tile; runs the full 2048-step encoder and
// 2048-step decoder locally. y ping-pongs between two LDS buffers; each wave32
// owns one 16-column output tile and drives v_wmma_f32_16x16x4_f32 over K=512.
__global__ __launch_bounds__(1024)
void lae_recurrence_kernel(const float* __restrict__ x,    // [128][2048][3]
                           const float* __restrict__ Aw,   // [512][3]
                           const float* __restrict__ Bw,   // [512][512]
                           const float* __restrict__ Cw,   // [515][512]
                           float* __restrict__ out)        // [128][2048][3]
{
  __shared__ float buf0[MTILE * STRIDE];
  __shared__ float buf1[MTILE * STRIDE];
  __shared__ float xs[MTILE * 4];

  const int tid  = threadIdx.x;
  const int wid  = tid >> 5;     // wave id 0..31
  const int lane = tid & 31;
  const int lo   = lane & 15;    // N column within tile (B/C/D frags), M row (A frag)
  const int hi   = lane >> 4;    // K half-select (A/B frags), M half-select (C/D frags)
  const int b0   = blockIdx.x * MTILE;

  // zero initial y state
  for (int i = tid; i < MTILE * STRIDE; i += 1024) buf0[i] = 0.0f;

  // ---------------- encoder: y_t = x_t A^T + y_{t-1} B^T ----------------
  const int n0 = wid * 16;  // this wave's 16 output columns
  // per-lane A_w coefficients for column n0+lo (folded into C init)
  const float a_c0 = Aw[(n0 + lo) * 3 + 0];
  const float a_c1 = Aw[(n0 + lo) * 3 + 1];
  const float a_c2 = Aw[(n0 + lo) * 3 + 2];
  // B fragment base: lane holds B_w[n0+lo][k + 2*hi .. +1]  (B^T(k,n) = B_w[n][k])
  const float* Brow = Bw + (size_t)(n0 + lo) * H_DIM + 2 * hi;

  auto enc_step = [&](const float* cur, float* nxt, int t) {
    if (tid < MTILE * 3) {                       // stage x_t tile (16x3)
      int b = tid / 3, d = tid - 3 * b;
      xs[b * 4 + d] = x[((size_t)(b0 + b) * T_SEQ + t) * 3 + d];
    }
    __syncthreads();                             // xs ready; prev-step writes visible

    v8f c;                                       // C init = x_t A^T (D layout: M = r + 8*hi, N = lo)
    #pragma unroll
    for (int r = 0; r < 8; ++r) {
      int m = r + 8 * hi;
      c[r] = xs[m * 4 + 0] * a_c0 + xs[m * 4 + 1] * a_c1 + xs[m * 4 + 2] * a_c2;
    }

    // A fragment: lane holds y[lo][k + 2*hi .. +1] from LDS (8B-aligned, bank-conflict-free)
    const float* yrow = cur + lo * STRIDE + YCOL + 2 * hi;
    #pragma unroll 4
    for (int k = 0; k < H_DIM; k += 4) {
      v2f a = *(const v2f*)(yrow + k);
      v2f b = *(const v2f*)(Brow + k);
      c = __builtin_amdgcn_wmma_f32_16x16x4_f32(false, a, false, b, (short)0, c,
                                                false, false);
    }

    // scatter D tile: lane writes rows r+8*hi, column n0+lo
    float* drow = nxt + (8 * hi) * STRIDE + YCOL + n0 + lo;
    #pragma unroll
    for (int r = 0; r < 8; ++r) drow[r * STRIDE] = c[r];
    __syncthreads();
  };

  for (int t = 0; t < T_SEQ; t += 2) {           // static ping-pong keeps LDS in addrspace(3)
    enc_step(buf0, buf1, t);
    enc_step(buf1, buf0, t + 1);
  }
  // final encoder state y is now in buf0 at columns YCOL..YCOL+511

  // ---------------- decoder: out = y C^T ; x_dec[:,i] = out[:,:3]; y = out[:,3:] ----------------
  // out has 515 columns -> 33 N-tiles; wave w owns tile w, wave 31 also owns tile 32.
  const bool extra = (wid == 31);
  const int nA  = 16 * wid + lo;                 // columns 0..511
  const int nBc = 512 + lo;                      // columns 512..527 (clamped for addressing)
  const float* CrowA = Cw + (size_t)nA * H_DIM + 2 * hi;
  const float* CrowB = Cw + (size_t)(nBc > 514 ? 514 : nBc) * H_DIM + 2 * hi;

  auto dec_step = [&](const float* cur, float* nxt, int i) {
    v8f c0 = {0.f,0.f,0.f,0.f,0.f,0.f,0.f,0.f};
    v8f c1 = {0.f,0.f,0.f,0.f,0.f,0.f,0.f,0.f};
    const float* yrow = cur + lo * STRIDE + YCOL + 2 * hi;  // y[k] = out_prev[3+k] at col 4+k
    if (extra) {                                 // wave-uniform branch: EXEC stays all-1s for WMMA
      #pragma unroll 2
      for (int k = 0; k < H_DIM; k += 4) {
        v2f a  = *(const v2f*)(yrow + k);
        v2f bA = *(const v2f*)(CrowA + k);
        v2f bB = *(const v2f*)(CrowB + k);
        c0 = __builtin_amdgcn_wmma_f32_16x16x4_f32(false, a, false, bA, (short)0, c0, false, false);
        c1 = __builtin_amdgcn_wmma_f32_16x16x4_f32(false, a, false, bB, (short)0, c1, false, false);
      }
    } else {
      #pragma unroll 4
      for (int k = 0; k < H_DIM; k += 4) {
        v2f a  = *(const v2f*)(yrow + k);
        v2f bA = *(const v2f*)(CrowA + k);
        c0 = __builtin_amdgcn_wmma_f32_16x16x4_f32(false, a, false, bA, (short)0, c0, false, false);
      }
    }

    // store out[n] at LDS column n+1 (so next y[k]=out[3+k] sits at col 4+k, aligned)
    float* drow = nxt + (8 * hi) * STRIDE + 1 + nA;
    #pragma unroll
    for (int r = 0; r < 8; ++r) drow[r * STRIDE] = c0[r];
    if (extra) {
      float* drow2 = nxt + (8 * hi) * STRIDE + 1 + nBc;    // cols 513..528 < STRIDE, pad-safe
      #pragma unroll
      for (int r = 0; r < 8; ++r) drow2[r * STRIDE] = c1[r];
    }
    // columns 0..2 of out -> x_dec[:, i, :]
    if (wid == 0 && lo < 3) {
      #pragma unroll
      for (int r = 0; r < 8; ++r) {
        int m = r + 8 * hi;
        out[((size_t)(b0 + m) * T_SEQ + i) * 3 + lo] = c0[r];
      }
    }
    __syncthreads();
  };

  for (int i = T_SEQ - 1; i > 0; i -= 2) {
    dec_step(buf0, buf1, i);
    dec_step(buf1, buf0, i - 1);
  }
}

extern "C" void kernel_launch(void* const* d_in, const int* in_sizes, int n_in,
                              void* d_out, int out_size, void* d_ws, size_t ws_size,
                              hipStream_t stream) {
  (void)in_sizes; (void)n_in; (void)out_size; (void)d_ws; (void)ws_size;
  const float* x  = (const float*)d_in[0];
  const float* Aw = (const float*)d_in[1];
  const float* Bw = (const float*)d_in[2];
  const float* Cw = (const float*)d_in[3];
  float* out = (float*)d_out;
  // 128 batch rows / 16 per workgroup = 8 workgroups, 1024 threads (32 wave32s) each
  lae_recurrence_kernel<<<dim3(128 / MTILE), dim3(1024), 0, stream>>>(x, Aw, Bw, Cw, out);
}